// CenterLoss_63015760167534
// MI455X (gfx1250) — compile-verified
//
#include <hip/hip_runtime.h>

#define NUM_CLASSES 6625
#define FEAT_DIM    96
#define N_ROWS      8192
#define CLIP_MIN    1e-12f
#define CLIP_MAX    1e12f

typedef __attribute__((ext_vector_type(2))) float v2f;
typedef __attribute__((ext_vector_type(8))) float v8f;

// ---------------------------------------------------------------------------
// Kernel 0: zero the global accumulator in workspace.
// ---------------------------------------------------------------------------
__global__ void cl_init_acc(float* __restrict__ acc) {
    if (threadIdx.x == 0) *acc = 0.0f;
}

// ---------------------------------------------------------------------------
// Kernel 1: row-wise argmax over predicts [N_ROWS, NUM_CLASSES].
// One 256-thread block per row; coalesced DWORD loads (row stride is only
// 4-byte aligned, so wider vector loads would be misaligned).
// Tie-break matches jnp.argmax: first occurrence of the maximum.
// ---------------------------------------------------------------------------
__global__ __launch_bounds__(256) void cl_argmax_rows(
        const float* __restrict__ pred, int* __restrict__ labels) {
    const int row = blockIdx.x;
    const float* __restrict__ p = pred + (size_t)row * NUM_CLASSES;
    const int t = threadIdx.x;

    float bestv = -3.402823466e38f;
    int   besti = NUM_CLASSES;           // larger than any valid index
    for (int i = t; i < NUM_CLASSES; i += 256) {
        float v = p[i];
        if (v > bestv) { bestv = v; besti = i; }   // strict > => first max
    }

    __shared__ float sv[256];
    __shared__ int   si[256];
    sv[t] = bestv;
    si[t] = besti;
    __syncthreads();
    #pragma unroll
    for (int s = 128; s > 0; s >>= 1) {
        if (t < s) {
            float ov = sv[t + s];
            int   oi = si[t + s];
            float mv = sv[t];
            if (ov > mv || (ov == mv && oi < si[t])) { sv[t] = ov; si[t] = oi; }
        }
        __syncthreads();
    }
    if (t == 0) labels[row] = si[0];
}

// ---------------------------------------------------------------------------
// Kernel 2: per-row squared distance via f32 WMMA.
// Each wave32 owns a 16-row tile. E = F_tile - C_gathered (16 x 96 f32).
// Accumulate E * E^T with V_WMMA_F32_16X16X4_F32 over 24 K-chunks; the
// diagonal of the 16x16 f32 accumulator is the per-row squared distance.
// E*E^T is symmetric, and the 16x4 A layout (lanes 0-15: K=0,1; lanes
// 16-31: K=2,3) makes the A and B register images identical, so the same
// VGPRs feed both operands. EXEC is all-ones through every WMMA.
// ---------------------------------------------------------------------------
__global__ __launch_bounds__(256) void cl_center_dist_wmma(
        const float* __restrict__ feats,
        const float* __restrict__ centers,
        const int*   __restrict__ labels,
        float*       __restrict__ acc) {
    const int lane   = threadIdx.x & 31;
    const int wave   = (blockIdx.x * blockDim.x + threadIdx.x) >> 5;
    const int row0   = wave * 16;
    const int rlocal = lane & 15;
    const int r      = row0 + rlocal;
    const int koff   = (lane >> 4) << 1;     // lanes 0-15 -> K+0, lanes 16-31 -> K+2

    const int lab = labels[r];
    const float* __restrict__ f = feats   + (size_t)r   * FEAT_DIM;
    const float* __restrict__ c = centers + (size_t)lab * FEAT_DIM;

    v8f accm = {0.f, 0.f, 0.f, 0.f, 0.f, 0.f, 0.f, 0.f};
    #pragma unroll
    for (int k0 = 0; k0 < FEAT_DIM; k0 += 4) {
        const int k = k0 + koff;
        v2f a;
        a.x = f[k]     - c[k];
        a.y = f[k + 1] - c[k + 1];
        // D = A * A^T + C   (8 args: neg_a, A, neg_b, B, c_mod, C, reuse_a, reuse_b)
        accm = __builtin_amdgcn_wmma_f32_16x16x4_f32(
                   false, a, false, a, (short)0, accm, false, false);
    }

    // Diagonal extraction per the 16x16 f32 C/D layout:
    //   VGPR v, lanes 0-15  -> (M=v,   N=lane)    => (v,v)     lives at lane v
    //   VGPR v, lanes 16-31 -> (M=v+8, N=lane-16) => (v+8,v+8) lives at lane v+24
    float dval = 0.0f;
    int   has  = 0;
    #pragma unroll
    for (int v = 0; v < 8; ++v) {
        if (lane == v)      { dval = accm[v]; has = 1; }
        if (lane == v + 24) { dval = accm[v]; has = 1; }
    }
    if (has) {
        dval = fminf(fmaxf(dval, CLIP_MIN), CLIP_MAX);
        atomicAdd(acc, dval);
    }
}

// ---------------------------------------------------------------------------
// Kernel 3: finalize -> (sum + N*(C-1)*1e-12) / N
// ---------------------------------------------------------------------------
__global__ void cl_finalize(const float* __restrict__ acc, float* __restrict__ out) {
    if (threadIdx.x == 0) {
        const float masked_off = (float)((double)N_ROWS * (double)(NUM_CLASSES - 1) * 1e-12);
        out[0] = (*acc + masked_off) / (float)N_ROWS;
    }
}

extern "C" void kernel_launch(void* const* d_in, const int* in_sizes, int n_in,
                              void* d_out, int out_size, void* d_ws, size_t ws_size,
                              hipStream_t stream) {
    const float* feats   = (const float*)d_in[0];   // [64,128,96]
    const float* pred    = (const float*)d_in[1];   // [64,128,6625]
    const float* centers = (const float*)d_in[2];   // [6625,96]
    float* out = (float*)d_out;

    float* acc    = (float*)d_ws;                       // 1 float
    int*   labels = (int*)((char*)d_ws + 256);          // 8192 ints

    cl_init_acc<<<1, 32, 0, stream>>>(acc);
    cl_argmax_rows<<<N_ROWS, 256, 0, stream>>>(pred, labels);
    // 8192 rows / 16 rows-per-wave = 512 waves; 8 waves per 256-thread block.
    cl_center_dist_wmma<<<(N_ROWS / 16) / 8, 256, 0, stream>>>(feats, centers, labels, acc);
    cl_finalize<<<1, 32, 0, stream>>>(acc, out);
}